// HATLayer_90177133346940
// MI455X (gfx1250) — compile-verified
//
#include <hip/hip_runtime.h>

#define D 128            // feature dim (reference uses 128)
#define WAVES_PER_BLOCK 8

// ---------------------------------------------------------------------------
// Edge scatter (placed FIRST so the disasm snippet shows the async gather
// loop). One edge per wave-iteration. Gather logmap[src] rows with CDNA5
// async global->LDS b128 copies, double buffered on ASYNCcnt, then
// global_atomic_add_f32 into segsum[dst] + count.
// ---------------------------------------------------------------------------
__global__ __launch_bounds__(256) void hat_scatter(const float* __restrict__ lm,
                                                   const long long* __restrict__ src,
                                                   const long long* __restrict__ dst,
                                                   float* __restrict__ segsum,
                                                   float* __restrict__ cnt,
                                                   long long E) {
    __shared__ float smem[WAVES_PER_BLOCK * 2 * D];   // 2 buffers x 512B per wave

    const int lane = threadIdx.x & 31;
    const int wv   = threadIdx.x >> 5;
    const long long nwaves = (long long)gridDim.x * WAVES_PER_BLOCK;
    long long e = (long long)blockIdx.x * WAVES_PER_BLOCK + wv;

    float* buf0 = &smem[wv * (2 * D)];
    float* buf1 = buf0 + D;
    // low 32 bits of the generic (flat) address of a __shared__ object are the
    // LDS byte offset -> exactly what the async instruction's VDST expects
    const unsigned loff0 = (unsigned)(size_t)(buf0 + lane * 4);
    const unsigned loff1 = (unsigned)(size_t)(buf1 + lane * 4);

    // prologue: kick off gather for first edge into buf0
    if (e < E) {
        const long long s = src[e];
        const float* g = lm + s * D + lane * 4;
        asm volatile("global_load_async_to_lds_b128 %0, %1, off"
                     :: "v"(loff0), "v"(g) : "memory");
    }

    int cur = 0;
    for (; e < E; e += nwaves) {
        const long long en = e + nwaves;
        if (en < E) {
            // issue next gather into the other buffer, then wait until only
            // that one is outstanding (current buffer complete)
            const long long sn = src[en];
            const float* gn = lm + sn * D + lane * 4;
            asm volatile("global_load_async_to_lds_b128 %0, %1, off"
                         :: "v"(cur ? loff0 : loff1), "v"(gn) : "memory");
            asm volatile("s_wait_asynccnt 0x1" ::: "memory");
        } else {
            asm volatile("s_wait_asynccnt 0x0" ::: "memory");
        }

        const float* buf = cur ? buf1 : buf0;
        const float4 v = *(const float4*)(buf + lane * 4);

        const long long d0 = dst[e];
        float* o = segsum + d0 * D + lane * 4;
        atomicAdd(o + 0, v.x);
        atomicAdd(o + 1, v.y);
        atomicAdd(o + 2, v.z);
        atomicAdd(o + 3, v.w);
        if (lane == 0) atomicAdd(cnt + d0, 1.0f);

        cur ^= 1;
    }
}

// ---------------------------------------------------------------------------
// Zero a float buffer (grid-stride, b128 stores + scalar tail)
// ---------------------------------------------------------------------------
__global__ __launch_bounds__(256) void hat_zero(float* __restrict__ p, long long n) {
    const long long n4 = n >> 2;
    long long i = (long long)blockIdx.x * blockDim.x + threadIdx.x;
    const long long stride = (long long)gridDim.x * blockDim.x;
    float4 z; z.x = 0.0f; z.y = 0.0f; z.z = 0.0f; z.w = 0.0f;
    for (long long j = i; j < n4; j += stride) ((float4*)p)[j] = z;
    for (long long j = (n4 << 2) + i; j < n; j += stride) p[j] = 0.0f;
}

// ---------------------------------------------------------------------------
// Per-node Poincare log-map at origin.
// One wave (32 lanes) per node; each lane owns 4 contiguous floats (b128).
// ---------------------------------------------------------------------------
__global__ __launch_bounds__(256) void hat_logmap(const float* __restrict__ feat,
                                                  float* __restrict__ lm, int N) {
    const int lane = threadIdx.x & 31;
    const int wv   = threadIdx.x >> 5;
    const int n    = blockIdx.x * WAVES_PER_BLOCK + wv;
    if (n >= N) return;

    const float4 v = *(const float4*)(feat + (long long)n * D + lane * 4);
    float ss = v.x * v.x + v.y * v.y + v.z * v.z + v.w * v.w;
    // wave32 butterfly reduction of sum-of-squares
    #pragma unroll
    for (int m = 16; m >= 1; m >>= 1) ss += __shfl_xor(ss, m, 32);

    const float norm = sqrtf(ss);
    float nc = fminf(fmaxf(norm, 1e-10f), 1.0f - 1e-5f);
    // artanh(nc) = 0.5*log((1+nc)/(1-nc)), via log1p for accuracy
    const float at    = 0.5f * log1pf(2.0f * nc / (1.0f - nc));
    const float scale = at / fmaxf(norm, 1e-10f);

    float4 o;
    o.x = v.x * scale; o.y = v.y * scale; o.z = v.z * scale; o.w = v.w * scale;
    *(float4*)(lm + (long long)n * D + lane * 4) = o;
}

// ---------------------------------------------------------------------------
// Per-node mean + Poincare exp-map at origin, in place on d_out.
// ---------------------------------------------------------------------------
__global__ __launch_bounds__(256) void hat_expmap(float* __restrict__ out,
                                                  const float* __restrict__ cnt, int N) {
    const int lane = threadIdx.x & 31;
    const int wv   = threadIdx.x >> 5;
    const int n    = blockIdx.x * WAVES_PER_BLOCK + wv;
    if (n >= N) return;

    const float inv = 1.0f / fmaxf(cnt[n], 1.0f);
    float4 v = *(const float4*)(out + (long long)n * D + lane * 4);
    v.x *= inv; v.y *= inv; v.z *= inv; v.w *= inv;

    float ss = v.x * v.x + v.y * v.y + v.z * v.z + v.w * v.w;
    #pragma unroll
    for (int m = 16; m >= 1; m >>= 1) ss += __shfl_xor(ss, m, 32);

    const float norm = sqrtf(ss);
    const float nc = fmaxf(norm, 1e-10f);
    const float scale = tanhf(nc) / nc;

    float4 o;
    o.x = v.x * scale; o.y = v.y * scale; o.z = v.z * scale; o.w = v.w * scale;
    *(float4*)(out + (long long)n * D + lane * 4) = o;
}

// ---------------------------------------------------------------------------
// Host launcher
// ---------------------------------------------------------------------------
extern "C" void kernel_launch(void* const* d_in, const int* in_sizes, int n_in,
                              void* d_out, int out_size, void* d_ws, size_t ws_size,
                              hipStream_t stream) {
    const float*     feat = (const float*)d_in[0];
    const long long* src  = (const long long*)d_in[1];
    const long long* dst  = (const long long*)d_in[2];

    const int       N = in_sizes[0] / D;      // 10000
    const long long E = in_sizes[1];          // 640000

    float* out = (float*)d_out;               // reused as segment-sum accumulator
    float* lm  = (float*)d_ws;                // logmap table: N*D floats
    float* cnt = lm + (size_t)N * D;          // counts: N floats

    // zero accumulators (harness does not re-poison between replays)
    hat_zero<<<1024, 256, 0, stream>>>(out, (long long)N * D);
    hat_zero<<<64,   256, 0, stream>>>(cnt, (long long)N);

    // precompute log-map per node (64x less work than per edge)
    hat_logmap<<<(N + WAVES_PER_BLOCK - 1) / WAVES_PER_BLOCK, 256, 0, stream>>>(feat, lm, N);

    // edge gather/scatter: ~20 edges per wave at 4096 blocks
    hat_scatter<<<4096, 256, 0, stream>>>(lm, src, dst, out, cnt, E);

    // mean + exp-map in place
    hat_expmap<<<(N + WAVES_PER_BLOCK - 1) / WAVES_PER_BLOCK, 256, 0, stream>>>(out, cnt, N);
}